// RandomMF_73254962200841
// MI455X (gfx1250) — compile-verified
//
#include <hip/hip_runtime.h>
#include <hip/hip_bf16.h>
#include <stdint.h>

#define IMG_H 256
#define IMG_W 256
#define TW 64
#define TH 32
#define PAD 2
#define LW (TW + 2 * PAD)   // 68
#define LH (TH + 2 * PAD)   // 36
#define LSTR 72             // padded LDS row stride (floats): 72 % 64 banks -> shift 8/row
#define NTHREADS 256

typedef __attribute__((address_space(1))) int g_int;  // global (AS1)
typedef __attribute__((address_space(3))) int l_int;  // LDS (AS3)

__device__ __forceinline__ int reflect_idx(int i, int n) {
    // jnp.pad mode='reflect' (no edge repeat): -1 -> 1, -2 -> 2, n -> n-2, n+1 -> n-3
    i = (i < 0) ? -i : i;
    i = (i >= n) ? (2 * n - 2 - i) : i;
    return i;
}

#define SW2(a, b) { float _t = fminf(a, b); b = fmaxf(a, b); a = _t; }

// Devillard opt_med25: 99 compare-exchange median-of-25 network (exact)
__device__ __forceinline__ float median25(float p[25]) {
    SW2(p[0], p[1]); SW2(p[3], p[4]); SW2(p[2], p[4]);
    SW2(p[2], p[3]); SW2(p[6], p[7]); SW2(p[5], p[7]);
    SW2(p[5], p[6]); SW2(p[9], p[10]); SW2(p[8], p[10]);
    SW2(p[8], p[9]); SW2(p[12], p[13]); SW2(p[11], p[13]);
    SW2(p[11], p[12]); SW2(p[15], p[16]); SW2(p[14], p[16]);
    SW2(p[14], p[15]); SW2(p[18], p[19]); SW2(p[17], p[19]);
    SW2(p[17], p[18]); SW2(p[21], p[22]); SW2(p[20], p[22]);
    SW2(p[20], p[21]); SW2(p[23], p[24]); SW2(p[2], p[5]);
    SW2(p[3], p[6]); SW2(p[0], p[6]); SW2(p[0], p[3]);
    SW2(p[4], p[7]); SW2(p[1], p[7]); SW2(p[1], p[4]);
    SW2(p[11], p[14]); SW2(p[8], p[14]); SW2(p[8], p[11]);
    SW2(p[12], p[15]); SW2(p[9], p[15]); SW2(p[9], p[12]);
    SW2(p[13], p[16]); SW2(p[10], p[16]); SW2(p[10], p[13]);
    SW2(p[20], p[23]); SW2(p[17], p[23]); SW2(p[17], p[20]);
    SW2(p[21], p[24]); SW2(p[18], p[24]); SW2(p[18], p[21]);
    SW2(p[19], p[22]); SW2(p[8], p[17]); SW2(p[9], p[18]);
    SW2(p[0], p[18]); SW2(p[0], p[9]); SW2(p[10], p[19]);
    SW2(p[1], p[19]); SW2(p[1], p[10]); SW2(p[11], p[20]);
    SW2(p[2], p[20]); SW2(p[2], p[11]); SW2(p[12], p[21]);
    SW2(p[3], p[21]); SW2(p[3], p[12]); SW2(p[13], p[22]);
    SW2(p[4], p[22]); SW2(p[4], p[13]); SW2(p[14], p[23]);
    SW2(p[5], p[23]); SW2(p[5], p[14]); SW2(p[15], p[24]);
    SW2(p[6], p[24]); SW2(p[6], p[15]); SW2(p[7], p[16]);
    SW2(p[7], p[19]); SW2(p[13], p[21]); SW2(p[15], p[23]);
    SW2(p[7], p[13]); SW2(p[7], p[15]); SW2(p[1], p[9]);
    SW2(p[3], p[11]); SW2(p[5], p[17]); SW2(p[11], p[17]);
    SW2(p[9], p[17]); SW2(p[4], p[10]); SW2(p[6], p[12]);
    SW2(p[7], p[14]); SW2(p[4], p[6]); SW2(p[4], p[7]);
    SW2(p[12], p[14]); SW2(p[10], p[14]); SW2(p[6], p[7]);
    SW2(p[10], p[12]); SW2(p[6], p[10]); SW2(p[6], p[17]);
    SW2(p[12], p[17]); SW2(p[7], p[17]); SW2(p[7], p[10]);
    SW2(p[12], p[18]); SW2(p[7], p[12]); SW2(p[10], p[18]);
    SW2(p[12], p[20]); SW2(p[10], p[20]); SW2(p[10], p[12]);
    return p[12];
}

__global__ __launch_bounds__(NTHREADS) void median5x5_kernel(
    const float* __restrict__ in, float* __restrict__ out) {
    __shared__ float tile[LH][LSTR];

    const int tx    = blockIdx.x;   // 0..3   tile col
    const int ty    = blockIdx.y;   // 0..7   tile row
    const int plane = blockIdx.z;   // 0..47  (B*C)
    const int tid   = threadIdx.x;

    const float* pin = in + (size_t)plane * (IMG_H * IMG_W);

    // ---- cooperative async load of the (68 x 36) halo tile into LDS ----
    // One division up front; then incremental (r,c): 256 = 3*68 + 52.
    const int total = LH * LW;  // 2448 elements
    int r = tid / LW;
    int c = tid - r * LW;
    const int row0 = ty * TH - PAD;
    const int col0 = tx * TW - PAD;
    for (int i = tid; i < total; i += NTHREADS) {
        const int gr = reflect_idx(row0 + r, IMG_H);
        const int gc = reflect_idx(col0 + c, IMG_W);
        const float* gp = pin + gr * IMG_W + gc;
        float* lp = &tile[r][c];
#if __has_builtin(__builtin_amdgcn_global_load_async_to_lds_b32)
        __builtin_amdgcn_global_load_async_to_lds_b32(
            (g_int*)(void*)gp, (l_int*)(void*)lp, 0, 0);
#else
        *lp = *gp;
#endif
        c += NTHREADS - 3 * LW;  // +52
        r += 3;
        if (c >= LW) { c -= LW; ++r; }
    }
#if __has_builtin(__builtin_amdgcn_s_wait_asynccnt)
    __builtin_amdgcn_s_wait_asynccnt(0);
#else
    asm volatile("s_wait_asynccnt 0" ::: "memory");
#endif
    __syncthreads();

    // ---- compute: each thread produces 8 output pixels (64x32 tile / 256 thr) ----
    // Thread (cx, qy) handles rows qy*8 .. qy*8+7 of the tile at column cx.
    // Full unroll lets the compiler keep overlapping LDS row-segments in VGPRs
    // (rows are shared by up to 5 consecutive windows).
    const int cx = tid & 63;
    const int qy = tid >> 6;
    float* pout = out + (size_t)plane * (IMG_H * IMG_W) + (ty * TH) * IMG_W + tx * TW;

#pragma unroll
    for (int j = 0; j < TH / 4; ++j) {
        const int rr = qy * (TH / 4) + j;   // 0..31
        float p[25];
#pragma unroll
        for (int dr = 0; dr < 5; ++dr) {
#pragma unroll
            for (int dc = 0; dc < 5; ++dc) {
                p[dr * 5 + dc] = tile[rr + dr][cx + dc];
            }
        }
        pout[rr * IMG_W + cx] = median25(p);
    }
}

extern "C" void kernel_launch(void* const* d_in, const int* in_sizes, int n_in,
                              void* d_out, int out_size, void* d_ws, size_t ws_size,
                              hipStream_t stream) {
    const float* image = (const float*)d_in[0];   // [16,3,256,256] f32
    // d_in[1] (cover_image) is unused by the reference computation.
    float* out = (float*)d_out;

    const int planes = in_sizes[0] / (IMG_H * IMG_W);  // B*C = 48
    dim3 grid(IMG_W / TW, IMG_H / TH, planes);
    median5x5_kernel<<<grid, NTHREADS, 0, stream>>>(image, out);
}